// MultiheadAttentionPooling_47562467836567
// MI455X (gfx1250) — compile-verified
//
#include <hip/hip_runtime.h>

// MultiheadAttentionPooling, algebraically reduced to 2 streaming passes over x.
// B=32, S=4096, C=768, H=12 (padded to 16), D=64.

#define Bn 32
#define Sn 4096
#define Cn 768
#define Hn 12
#define HP 16
#define Dn 64

typedef __attribute__((ext_vector_type(2))) float v2f;
typedef __attribute__((ext_vector_type(8))) float v8f;

// workspace layout (float offsets)
#define WKB_OFF   0u                               // (C/4)*64 = 12288  (B-operand-shuffled wk_eff)
#define SB_OFF    12288u                           // 16
#define SC_OFF    12352u                           // B*S*HP = 2097152  (scores, then attn, [b][s][h16])
#define XBP_OFF   (12352u + 2097152u)              // 4*B*HP*C = 1572864 (split-K partial xbar)
#define CTX_OFF   (12352u + 2097152u + 1572864u)   // B*C = 24576

// ---------------- kernel 0: rank-1 precompute ------------------------------
// qs[h,d] = (probe . Wq[:,h,d] + bq) / sqrt(D)
// wk_eff[c,h] = sum_d Wk[c,h,d]*qs[h,d]   -> stored pre-shuffled into WMMA B layout
// sbias[h] = sum_d bk[h,d]*qs[h,d]
__global__ void __launch_bounds__(256) prep_kernel(const float* __restrict__ probe,
    const float* __restrict__ Wq, const float* __restrict__ bq,
    const float* __restrict__ Wk, const float* __restrict__ bk,
    float* __restrict__ wkB, float* __restrict__ sbias)
{
    __shared__ float probeL[Cn];
    __shared__ float qsL[Cn];        // H*D == C == 768
    int tid = threadIdx.x;
    for (int i = tid; i < Cn; i += 256) probeL[i] = probe[i];
    __syncthreads();
    for (int i = tid; i < Cn; i += 256) {            // i = h*64+d
        float acc = bq[i];
        for (int c = 0; c < Cn; ++c) acc = fmaf(probeL[c], Wq[(size_t)c * Cn + i], acc);
        qsL[i] = acc * 0.125f;                        // 1/sqrt(64)
    }
    __syncthreads();
    if (tid < HP) {
        float s = 0.f;
        if (tid < Hn)
            for (int d = 0; d < Dn; ++d) s = fmaf(bk[tid * Dn + d], qsL[tid * Dn + d], s);
        sbias[tid] = s;
    }
    for (int idx = tid; idx < Cn * HP; idx += 256) {
        int c = idx >> 4, h = idx & 15;
        float v = 0.f;
        if (h < Hn)
            for (int d = 0; d < Dn; ++d) v = fmaf(Wk[(size_t)c * Cn + h * Dn + d], qsL[h * Dn + d], v);
        // WMMA B-operand layout for K-block c>>2: lane holds (N=h, Kpair=(lane>>4)*2)
        int lane = (((c >> 1) & 1) << 4) + h;
        wkB[(c >> 2) * 64 + lane * 2 + (c & 1)] = v;
    }
}

// ---------------- kernel 1: scores = x @ wk_eff  (WMMA f32 16x16x4) --------
// one wave -> 16(s) x 16(h) tile, K=768; output scoresT[b][s][h16]
__global__ void __launch_bounds__(256) scores_kernel(const float* __restrict__ x,
    const float* __restrict__ wkB, const float* __restrict__ sbias,
    float* __restrict__ scoresT)
{
    __shared__ float wkL[(Cn / 4) * 64];   // 48 KB
    __shared__ float sbL[HP];
    int tid = threadIdx.x;
    for (int i = tid; i < (Cn / 4) * 64; i += 256) wkL[i] = wkB[i];
    if (tid < HP) sbL[tid] = sbias[tid];
    __syncthreads();

    int wave = tid >> 5, lane = tid & 31;
    int tile = blockIdx.x * 8 + wave;          // 0..8191 = B*S/16
    int b  = tile >> 8;                        // 256 tiles per batch
    int st = (tile & 255) << 4;                // s base
    int row  = lane & 15;
    int koff = (lane >> 4) << 1;               // A layout: VGPR pair holds K = koff, koff+1
    const float* xrow = x + ((size_t)b * Sn + st + row) * Cn + koff;

    v8f acc = {};
    for (int k = 0; k < Cn; k += 4) {
        v2f a  = *(const v2f*)(xrow + k);
        v2f bb = *(const v2f*)(&wkL[(k >> 2) * 64 + (lane << 1)]);
        acc = __builtin_amdgcn_wmma_f32_16x16x4_f32(false, a, false, bb,
                                                    (short)0, acc, false, false);
    }
    int h  = lane & 15;
    float sb = sbL[h];
    int m0 = (lane >> 4) << 3;                 // D layout: VGPR r -> M = m0 + r
    float* outp = scoresT + ((size_t)b * Sn + st + m0) * HP + h;
#pragma unroll
    for (int r = 0; r < 8; ++r) outp[(size_t)r * HP] = acc[r] + sb;
}

// ---------------- kernel 2: softmax over s, in place -----------------------
__global__ void __launch_bounds__(256) softmax_kernel(float* __restrict__ scoresT)
{
    __shared__ float red[256];
    int b = blockIdx.x >> 4, h = blockIdx.x & 15;
    int tid = threadIdx.x;
    float* p = scoresT + (size_t)b * Sn * HP + h;
    float v[16];
    float m = -3.4e38f;
#pragma unroll
    for (int j = 0; j < 16; ++j) { v[j] = p[(size_t)(tid + j * 256) * HP]; m = fmaxf(m, v[j]); }
    red[tid] = m; __syncthreads();
    for (int s = 128; s > 0; s >>= 1) { if (tid < s) red[tid] = fmaxf(red[tid], red[tid + s]); __syncthreads(); }
    m = red[0]; __syncthreads();
    float sum = 0.f;
#pragma unroll
    for (int j = 0; j < 16; ++j) { v[j] = __expf(v[j] - m); sum += v[j]; }
    red[tid] = sum; __syncthreads();
    for (int s = 128; s > 0; s >>= 1) { if (tid < s) red[tid] += red[tid + s]; __syncthreads(); }
    float inv = 1.0f / red[0];
#pragma unroll
    for (int j = 0; j < 16; ++j) p[(size_t)(tid + j * 256) * HP] = v[j] * inv;
}

// ---------------- kernel 3: xbar = attn @ x  (WMMA f32 16x16x4, split-K) ---
// one wave -> 16(h) x 16(c) tile over an S-chunk of 1024; x read exactly once
__global__ void __launch_bounds__(256) xbar_kernel(const float* __restrict__ x,
    const float* __restrict__ attnT, float* __restrict__ xbarp)
{
    int wg = blockIdx.x;
    int cg = wg % 6;
    int b  = (wg / 6) & 31;
    int chunk = wg / (6 * 32);                 // 0..3
    int tid = threadIdx.x;
    int wave = tid >> 5, lane = tid & 31;
    int c  = cg * 128 + wave * 16 + (lane & 15);
    int kb = (lane >> 4) << 1;
    const float* ap = attnT + ((size_t)b * Sn + chunk * 1024 + kb) * HP + (lane & 15);
    const float* xp = x     + ((size_t)b * Sn + chunk * 1024 + kb) * Cn + c;

    v8f acc = {};
    for (int s = 0; s < 1024; s += 4) {
        v2f a, bb;
        a.x  = ap[(size_t)s * HP];
        a.y  = ap[(size_t)(s + 1) * HP];
        bb.x = xp[(size_t)s * Cn];
        bb.y = xp[(size_t)(s + 1) * Cn];
        acc = __builtin_amdgcn_wmma_f32_16x16x4_f32(false, a, false, bb,
                                                    (short)0, acc, false, false);
    }
    int m0 = (lane >> 4) << 3;
    float* op = xbarp + (((size_t)chunk * Bn + b) * HP + m0) * Cn + c;
#pragma unroll
    for (int r = 0; r < 8; ++r) op[(size_t)r * Cn] = acc[r];
}

// ---------------- kernel 4: ctx[b,h,d] = xbar[b,h,:] @ Wv[:,h,d] + bv ------
__global__ void __launch_bounds__(64) ctx_kernel(const float* __restrict__ xbarp,
    const float* __restrict__ Wv, const float* __restrict__ bv, float* __restrict__ ctx)
{
    __shared__ float xs[Cn];
    int b = blockIdx.x / Hn, h = blockIdx.x % Hn;
    int tid = threadIdx.x;
    for (int i = tid; i < Cn; i += 64) {
        float s = 0.f;
#pragma unroll
        for (int k = 0; k < 4; ++k) s += xbarp[(((size_t)k * Bn + b) * HP + h) * Cn + i];
        xs[i] = s;                              // split-K reduce
    }
    __syncthreads();
    float acc = bv[h * Dn + tid];
    for (int cc = 0; cc < Cn; ++cc)
        acc = fmaf(xs[cc], Wv[(size_t)cc * Cn + h * Dn + tid], acc);
    ctx[(size_t)b * Cn + h * Dn + tid] = acc;
}

// ---------------- kernel 5: out[b,:] = ctx[b,:] @ Wo + bo ------------------
__global__ void __launch_bounds__(256) out_kernel(const float* __restrict__ ctx,
    const float* __restrict__ Wo, const float* __restrict__ bo, float* __restrict__ out)
{
    __shared__ float cl[Cn];
    int b = blockIdx.x, tid = threadIdx.x;
    for (int i = tid; i < Cn; i += 256) cl[i] = ctx[(size_t)b * Cn + i];
    __syncthreads();
    for (int cc = tid; cc < Cn; cc += 256) {
        float acc = bo[cc];
        for (int hd = 0; hd < Cn; ++hd)       // H*D == 768 == C
            acc = fmaf(cl[hd], Wo[(size_t)hd * Cn + cc], acc);
        out[(size_t)b * Cn + cc] = acc;
    }
}

extern "C" void kernel_launch(void* const* d_in, const int* in_sizes, int n_in,
                              void* d_out, int out_size, void* d_ws, size_t ws_size,
                              hipStream_t stream) {
    const float* x     = (const float*)d_in[0];
    const float* probe = (const float*)d_in[1];
    const float* Wq    = (const float*)d_in[2];
    const float* bq    = (const float*)d_in[3];
    const float* Wk    = (const float*)d_in[4];
    const float* bk    = (const float*)d_in[5];
    const float* Wv    = (const float*)d_in[6];
    const float* bv    = (const float*)d_in[7];
    const float* Wo    = (const float*)d_in[8];
    const float* bo    = (const float*)d_in[9];
    float* out = (float*)d_out;
    float* ws  = (float*)d_ws;

    float* wkB     = ws + WKB_OFF;
    float* sbias   = ws + SB_OFF;
    float* scoresT = ws + SC_OFF;
    float* xbarp   = ws + XBP_OFF;
    float* ctx     = ws + CTX_OFF;

    hipLaunchKernelGGL(prep_kernel,   dim3(1),          dim3(256), 0, stream, probe, Wq, bq, Wk, bk, wkB, sbias);
    hipLaunchKernelGGL(scores_kernel, dim3(1024),       dim3(256), 0, stream, x, wkB, sbias, scoresT);
    hipLaunchKernelGGL(softmax_kernel,dim3(Bn * HP),    dim3(256), 0, stream, scoresT);
    hipLaunchKernelGGL(xbar_kernel,   dim3(4 * Bn * 6), dim3(256), 0, stream, x, scoresT, xbarp);
    hipLaunchKernelGGL(ctx_kernel,    dim3(Bn * Hn),    dim3(64),  0, stream, xbarp, Wv, bv, ctx);
    hipLaunchKernelGGL(out_kernel,    dim3(Bn),         dim3(256), 0, stream, ctx, Wo, bo, out);
}